// NetworkClass_31018253812098
// MI455X (gfx1250) — compile-verified
//
#include <hip/hip_runtime.h>

typedef __attribute__((ext_vector_type(2))) float v2f;
typedef __attribute__((ext_vector_type(8))) float v8f;

#define NMAT  2048
#define KTOT  4096
#define ALPHA 0.995f
#define BETA  0.975f
#define VTH   2.0f

#define BM 128
#define BN 128
#define BK 32

// ---- CDNA5 async global->LDS path (ASYNCcnt), with portable fallback -------
#if __has_builtin(__builtin_amdgcn_global_load_async_to_lds_b128)
#define USE_ASYNC 1
#else
#define USE_ASYNC 0
#endif

// Builtin signature (from hipcc diagnostic): arg0 is AS(1) pointer to
// 'int __attribute__((vector_size(16)))'. Use exactly that pointee type.
typedef int v4i_vs __attribute__((vector_size(16)));
typedef __attribute__((address_space(1))) v4i_vs gas_v4i;  // global int4
typedef __attribute__((address_space(3))) v4i_vs las_v4i;  // LDS int4

__device__ __forceinline__ void async_copy16(const float* g, float* l) {
#if USE_ASYNC
    __builtin_amdgcn_global_load_async_to_lds_b128((gas_v4i*)g, (las_v4i*)l, 0, 0);
#else
    *(float4*)l = *(const float4*)g;
#endif
}

__device__ __forceinline__ void pipeline_barrier() {
#if USE_ASYNC
#if __has_builtin(__builtin_amdgcn_s_wait_asynccnt)
    __builtin_amdgcn_s_wait_asynccnt(0);
#else
    asm volatile("s_wait_asynccnt 0x0" ::: "memory");
#endif
#endif
    __syncthreads();
}

// ---------------------------------------------------------------------------
// Kernel 1: z_out_new = BETA * z_out + z  (d_out slice 2; also GEMM A input)
// ---------------------------------------------------------------------------
__global__ __launch_bounds__(256) void zo_kernel(const float* __restrict__ z,
                                                 const float* __restrict__ zout,
                                                 float* __restrict__ ZO) {
    int i = blockIdx.x * blockDim.x + threadIdx.x;  // float4 index
    float4 a = ((const float4*)zout)[i];
    float4 b = ((const float4*)z)[i];
    float4 o;
    o.x = BETA * a.x + b.x;
    o.y = BETA * a.y + b.y;
    o.z = BETA * a.z + b.z;
    o.w = BETA * a.w + b.w;
    ((float4*)ZO)[i] = o;
}

// ---------------------------------------------------------------------------
// Kernel 2: fused GEMM, K = 4096:
//   v_new = [X | ZO] @ [W ; WREC] + ALPHA*v - VTH*z
// fp32 WMMA, block tile 128x128, 8 wave32s, 32x64 per wave (2x4 accs),
// double-buffered LDS fed by GLOBAL_LOAD_ASYNC_TO_LDS_B128.
// A tile stored with XOR-swizzled 16B groups: aligned async stores +
// low-conflict fragment reads without padding (keeps LDS at exactly 64KB).
// ---------------------------------------------------------------------------
__global__ __launch_bounds__(256) void lif_gemm_kernel(
    const float* __restrict__ X, const float* __restrict__ W,
    const float* __restrict__ ZO, const float* __restrict__ WR,
    const float* __restrict__ V, const float* __restrict__ Z,
    float* __restrict__ OutV) {
    __shared__ float As[2][BM][BK];  // [m][swizzled k]  16KB per buffer
    __shared__ float Bs[2][BK][BN];  // [k][n]           16KB per buffer

    const int tid  = threadIdx.x;
    const int lane = tid & 31;
    const int wave = tid >> 5;
    const int wm   = wave & 3;   // 0..3 : wave row (32 rows each)
    const int wn   = wave >> 2;  // 0..1 : wave col (64 cols each)
    const int hl   = lane >> 4;  // K-pair select (lane half)
    const int lr   = lane & 15;  // row/col within 16

    const int m0 = blockIdx.y * BM;
    const int n0 = blockIdx.x * BN;

    v8f acc[2][4] = {};

    // --- stage one BKxBM / BKxBN tile pair into LDS buffer `buf` -----------
    auto stage = [&](int buf, int kt) {
        const float* Ag = (kt < 2048) ? X : ZO;
        const float* Bg = (kt < 2048) ? W : WR;
        const int k0 = kt & 2047;
        // A tile: 128 rows x 32 cols, 4 x b128 per thread, swizzled dest
        #pragma unroll
        for (int it = 0; it < 4; ++it) {
            int idx = tid + it * 256;
            int r   = idx >> 3;                 // 0..127
            int c4  = idx & 7;                  // 16B group 0..7
            int cs  = (c4 ^ (r & 7)) * 4;       // XOR swizzle
            async_copy16(Ag + (size_t)(m0 + r) * NMAT + k0 + c4 * 4,
                         &As[buf][r][cs]);
        }
        // B tile: 32 rows x 128 cols, 4 x b128 per thread, row-major dest
        #pragma unroll
        for (int it = 0; it < 4; ++it) {
            int idx = tid + it * 256;
            int r   = idx >> 5;                 // 0..31
            int c   = (idx & 31) * 4;           // 0..124
            async_copy16(Bg + (size_t)(k0 + r) * NMAT + n0 + c,
                         &Bs[buf][r][c]);
        }
    };

    // --- 64 WMMAs on LDS buffer `buf` --------------------------------------
    auto compute = [&](int buf) {
        #pragma unroll
        for (int kk = 0; kk < BK; kk += 4) {
            v2f afrag[2];
            v2f bfrag[4];
            #pragma unroll
            for (int mi = 0; mi < 2; ++mi) {
                int m  = wm * 32 + mi * 16 + lr;
                int cs = (((kk >> 2) ^ (lr & 7)) * 4) + 2 * hl;  // swizzled col
                afrag[mi] = *(const v2f*)&As[buf][m][cs];
            }
            #pragma unroll
            for (int nj = 0; nj < 4; ++nj) {
                int c = wn * 64 + nj * 16 + lr;
                bfrag[nj].x = Bs[buf][kk + 2 * hl + 0][c];
                bfrag[nj].y = Bs[buf][kk + 2 * hl + 1][c];
            }
            #pragma unroll
            for (int mi = 0; mi < 2; ++mi)
                #pragma unroll
                for (int nj = 0; nj < 4; ++nj)
                    acc[mi][nj] = __builtin_amdgcn_wmma_f32_16x16x4_f32(
                        false, afrag[mi], false, bfrag[nj], (short)0,
                        acc[mi][nj], false, false);
        }
    };

    // --- software pipeline: stage k+1 while computing k --------------------
    stage(0, 0);
    pipeline_barrier();
    int cur = 0;
    for (int kt = 0; kt < KTOT; kt += BK) {
        if (kt + BK < KTOT) stage(cur ^ 1, kt + BK);  // uniform branch
        compute(cur);
        pipeline_barrier();
        cur ^= 1;
    }

    // --- epilogue: + ALPHA*v - VTH*z ---------------------------------------
    #pragma unroll
    for (int mi = 0; mi < 2; ++mi) {
        #pragma unroll
        for (int nj = 0; nj < 4; ++nj) {
            int col = n0 + wn * 64 + nj * 16 + lr;
            #pragma unroll
            for (int r = 0; r < 8; ++r) {
                int row    = m0 + wm * 32 + mi * 16 + r + hl * 8;
                size_t off = (size_t)row * NMAT + col;
                OutV[off]  = acc[mi][nj][r] + ALPHA * V[off] - VTH * Z[off];
            }
        }
    }
}

// ---------------------------------------------------------------------------
// Kernel 3: z_new[i][j] = (v_new[0][i] > VTH) ? 1 : 0   (row broadcast)
// ---------------------------------------------------------------------------
__global__ __launch_bounds__(256) void mask_kernel(const float* __restrict__ Vout,
                                                   float* __restrict__ Znew) {
    int row = blockIdx.x;
    float m = (Vout[row] - VTH > 0.0f) ? 1.0f : 0.0f;  // Vout[row] == v_new[0][row]
    float4 mv = make_float4(m, m, m, m);
    float4* dst = (float4*)(Znew + (size_t)row * NMAT);
    for (int j = threadIdx.x; j < NMAT / 4; j += blockDim.x) dst[j] = mv;
}

// ---------------------------------------------------------------------------
extern "C" void kernel_launch(void* const* d_in, const int* in_sizes, int n_in,
                              void* d_out, int out_size, void* d_ws, size_t ws_size,
                              hipStream_t stream) {
    const float* x    = (const float*)d_in[0];
    const float* v    = (const float*)d_in[1];
    const float* z    = (const float*)d_in[2];
    const float* zout = (const float*)d_in[3];
    const float* w    = (const float*)d_in[4];
    const float* wrec = (const float*)d_in[5];

    float* outV  = (float*)d_out;                   // v_new     [2048x2048]
    float* outZ  = outV + (size_t)NMAT * NMAT;      // z_new     [2048x2048]
    float* outZO = outV + 2 * (size_t)NMAT * NMAT;  // z_out_new [2048x2048]

    // 1) z_out_new (also GEMM input)
    zo_kernel<<<(NMAT * NMAT / 4) / 256, 256, 0, stream>>>(z, zout, outZO);

    // 2) fused K=4096 fp32 WMMA GEMM + epilogue
    dim3 grid(NMAT / BN, NMAT / BM);  // (16, 16)
    lif_gemm_kernel<<<grid, 256, 0, stream>>>(x, w, outZO, wrec, v, z, outV);

    // 3) threshold row-0 of v_new, broadcast across rows of z_new
    mask_kernel<<<NMAT, 256, 0, stream>>>(outV, outZ);
}